// AttentionGroupPool_25915832664145
// MI455X (gfx1250) — compile-verified
//
#include <hip/hip_runtime.h>
#include <hip/hip_bf16.h>
#include <cstdint>

// AttentionGroupPool for MI455X (gfx1250).
// Bandwidth-bound (134 MB read @ 23.3 TB/s ~= 5.8 us floor); uses the CDNA5
// async global->LDS copy path (ASYNCcnt) + wave32 shuffle softmax.

#define B_ROWS   8192
#define T_COLS   4096
#define NG       64
#define NTHREADS 256   // one 16-float chunk per thread; 16*256 = 4096 cols

__global__ __launch_bounds__(NTHREADS)
void attn_group_pool_kernel(const float* __restrict__ H,
                            const float* __restrict__ w_ptr,
                            const float* __restrict__ b_ptr,
                            float* __restrict__ Z,      // [B]
                            float* __restrict__ A)      // [B, 64]
{
    __shared__ __align__(16) float sH[T_COLS];     // 16 KiB row tile
    __shared__ float sPart[NTHREADS];              // per-chunk partial sums

    const int tid = threadIdx.x;
    const int row = blockIdx.x;

    // ---- Stage this row into LDS via CDNA5 async global->LDS b128 copies ----
    // Each lane copies its own 64B chunk; since a lane reads back only the
    // bytes it issued, a per-wave s_wait_asynccnt suffices (no barrier yet).
    {
        const uint64_t gbase  = (uint64_t)(H + (size_t)row * T_COLS);
        const uint32_t ldsAdr = (uint32_t)(uintptr_t)(&sH[0]) + (uint32_t)tid * 64u;
        const uint32_t gOff   = (uint32_t)tid * 64u;
        asm volatile(
            "global_load_async_to_lds_b128 %0, %1, %2 offset:0\n\t"
            "global_load_async_to_lds_b128 %0, %1, %2 offset:16\n\t"
            "global_load_async_to_lds_b128 %0, %1, %2 offset:32\n\t"
            "global_load_async_to_lds_b128 %0, %1, %2 offset:48"
            :
            : "v"(ldsAdr), "v"(gOff), "s"(gbase)
            : "memory");
        asm volatile("s_wait_asynccnt 0x0" ::: "memory");
    }

    // ---- Per-chunk partial sum: 4x ds_load_b128, 16-wide add tree ----
    {
        const float4* c4 = (const float4*)(sH + tid * 16);
        float4 v0 = c4[0], v1 = c4[1], v2 = c4[2], v3 = c4[3];
        float ps = ((v0.x + v0.y) + (v0.z + v0.w))
                 + ((v1.x + v1.y) + (v1.z + v1.w))
                 + ((v2.x + v2.y) + (v2.z + v2.w))
                 + ((v3.x + v3.y) + (v3.z + v3.w));
        sPart[tid] = ps;
    }
    __syncthreads();

    // ---- Wave 0 finishes: 2 groups per lane, wave32 shuffle softmax ----
    if (tid < 32) {
        const float w  = w_ptr[0];
        const float bb = b_ptr[0];
        // per 256-col pattern block (16 chunks): group sub-layout
        const int   startLut[4] = {0, 2, 6, 12};
        const int   cntLut[4]   = {2, 4, 6, 4};
        const float invSz[4]    = {1.0f/32.0f, 1.0f/64.0f, 1.0f/96.0f, 1.0f/64.0f};

        float Gv[2], Sv[2];
#pragma unroll
        for (int h = 0; h < 2; ++h) {
            const int g    = tid + 32 * h;
            const int p    = g >> 2;
            const int sub  = g & 3;
            const int base = (p << 4) + startLut[sub];
            float s = 0.0f;
            for (int j = 0; j < cntLut[sub]; ++j) s += sPart[base + j];
            Gv[h] = s * invSz[sub];
            Sv[h] = fmaf(Gv[h], w, bb);
        }

        // numerically-stable softmax over 64 entries held as 2-per-lane
        float m = fmaxf(Sv[0], Sv[1]);
#pragma unroll
        for (int off = 16; off > 0; off >>= 1)
            m = fmaxf(m, __shfl_xor(m, off, 32));

        const float e0 = __expf(Sv[0] - m);
        const float e1 = __expf(Sv[1] - m);
        float den = e0 + e1;
        float num = e0 * Gv[0] + e1 * Gv[1];
#pragma unroll
        for (int off = 16; off > 0; off >>= 1) {
            den += __shfl_xor(den, off, 32);
            num += __shfl_xor(num, off, 32);
        }
        const float inv = 1.0f / den;

        float* aRow = A + (size_t)row * NG;
        aRow[tid]      = e0 * inv;
        aRow[tid + 32] = e1 * inv;
        if (tid == 0) Z[row] = num * inv;
    }
}

extern "C" void kernel_launch(void* const* d_in, const int* in_sizes, int n_in,
                              void* d_out, int out_size, void* d_ws, size_t ws_size,
                              hipStream_t stream) {
    (void)in_sizes; (void)n_in; (void)out_size; (void)d_ws; (void)ws_size;
    const float* H  = (const float*)d_in[0];   // [B, T] f32
    const float* sw = (const float*)d_in[1];   // [1,1]
    const float* sb = (const float*)d_in[2];   // [1]
    float* Z = (float*)d_out;                  // [B]
    float* A = (float*)d_out + B_ROWS;         // [B, 64]

    dim3 grid(B_ROWS);
    dim3 block(NTHREADS);
    attn_group_pool_kernel<<<grid, block, 0, stream>>>(H, sw, sb, Z, A);
}